// GraphGather_39771397161031
// MI455X (gfx1250) — compile-verified
//
#include <hip/hip_runtime.h>
#include <hip/hip_bf16.h>

// Problem constants from the reference: D = 256, segments contiguous in rows.
#define SEG_D 256

typedef __attribute__((ext_vector_type(2))) float v2f;
typedef __attribute__((ext_vector_type(8))) float v8f;

// ---------------------------------------------------------------------------
// Kernel 1: exclusive prefix-sum of the per-segment sizes -> offsets in d_ws.
// One block of 1024 threads; per-thread serial chunk + LDS Hillis-Steele.
// ---------------------------------------------------------------------------
__global__ __launch_bounds__(1024) void seg_offsets_kernel(
    const int* __restrict__ sizes, int* __restrict__ offsets, int B) {
  __shared__ int tmp[1024];
  const int t = threadIdx.x;
  const int chunk = (B + 1023) / 1024;
  const int base = t * chunk;

  int s = 0;
  for (int i = 0; i < chunk; ++i) {
    int idx = base + i;
    if (idx < B) s += sizes[idx];
  }
  tmp[t] = s;
  __syncthreads();

  for (int off = 1; off < 1024; off <<= 1) {
    int v = (t >= off) ? tmp[t - off] : 0;
    __syncthreads();
    tmp[t] += v;
    __syncthreads();
  }

  int run = (t == 0) ? 0 : tmp[t - 1];  // exclusive prefix of this chunk
  for (int i = 0; i < chunk; ++i) {
    int idx = base + i;
    if (idx < B) {
      offsets[idx] = run;
      run += sizes[idx];
    }
  }
  if (t == 1023) offsets[B] = tmp[1023];
}

// ---------------------------------------------------------------------------
// Kernel 2: segment mean. One block per segment, 512 threads = 16 wave32s.
// Wave w reduces the 16-column tile [16w, 16w+16) on the matrix pipe:
// V_WMMA_F32_16X16X4_F32 with A = ones(16x4) gives D[m,n] = sum_k B[k,n] + C,
// i.e. each WMMA folds 4 rows x 16 cols of x into the accumulator.
// B layout (32-bit, 4x16): VGPR v, lane l holds B[k = 2v + (l>>4)][n = l&15].
// Main loop: unconditional NT loads, immediate offsets, software-pipelined.
// Tail (<8 rows): clamped addresses + select-mask -> branch-free, EXEC all-1s.
// ---------------------------------------------------------------------------
__global__ __launch_bounds__(512) void seg_mean_wmma_kernel(
    const float* __restrict__ x, const int* __restrict__ sizes,
    const int* __restrict__ offsets, float* __restrict__ out) {
  const int b = blockIdx.x;
  const int start = offsets[b];
  const int size  = sizes[b];
  const int end   = start + size;

  const int wave = threadIdx.x >> 5;   // 0..15 -> column tile
  const int lane = threadIdx.x & 31;
  const int n    = lane & 15;          // column within tile
  const int kh   = lane >> 4;          // 0/1: which K row this half-wave holds
  const int col  = (wave << 4) + n;

  const float* __restrict__ xc = x + col;

  v2f a;  a.x = 1.0f;  a.y = 1.0f;     // A = ones(16x4)
  v8f acc0 = {};
  v8f acc1 = {};

  // ---- main loop: full 8-row chunks, no predication --------------------
  const int nchunks = size >> 3;
  const float* p = xc + (long)(start + kh) * SEG_D;

  if (nchunks > 0) {
    v2f b0, b1;
    b0.x = __builtin_nontemporal_load(p + 0 * SEG_D);
    b0.y = __builtin_nontemporal_load(p + 2 * SEG_D);
    b1.x = __builtin_nontemporal_load(p + 4 * SEG_D);
    b1.y = __builtin_nontemporal_load(p + 6 * SEG_D);

    for (int i = 1; i < nchunks; ++i) {
      p += 8 * SEG_D;
      __builtin_prefetch(p + 8 * SEG_D, 0, 1);  // global_prefetch_b8, look-ahead
      v2f c0, c1;
      c0.x = __builtin_nontemporal_load(p + 0 * SEG_D);
      c0.y = __builtin_nontemporal_load(p + 2 * SEG_D);
      c1.x = __builtin_nontemporal_load(p + 4 * SEG_D);
      c1.y = __builtin_nontemporal_load(p + 6 * SEG_D);
      acc0 = __builtin_amdgcn_wmma_f32_16x16x4_f32(
          false, a, false, b0, (short)0, acc0, false, false);
      acc1 = __builtin_amdgcn_wmma_f32_16x16x4_f32(
          false, a, false, b1, (short)0, acc1, false, false);
      b0 = c0;
      b1 = c1;
    }
    acc0 = __builtin_amdgcn_wmma_f32_16x16x4_f32(
        false, a, false, b0, (short)0, acc0, false, false);
    acc1 = __builtin_amdgcn_wmma_f32_16x16x4_f32(
        false, a, false, b1, (short)0, acc1, false, false);
  }

  // ---- tail: remaining 1..7 rows, branch-free masking ------------------
  const int done = start + (nchunks << 3);
  if (done < end) {                    // uniform across the wave
    const int last = end - 1;
    int r0 = done + 0 + kh;
    int r1 = done + 2 + kh;
    int r2 = done + 4 + kh;
    int r3 = done + 6 + kh;
    float m0 = (r0 <= last) ? 1.0f : 0.0f;
    float m1 = (r1 <= last) ? 1.0f : 0.0f;
    float m2 = (r2 <= last) ? 1.0f : 0.0f;
    float m3 = (r3 <= last) ? 1.0f : 0.0f;
    int c0i = (r0 <= last) ? r0 : last;  // clamped -> always a valid address
    int c1i = (r1 <= last) ? r1 : last;
    int c2i = (r2 <= last) ? r2 : last;
    int c3i = (r3 <= last) ? r3 : last;

    v2f b0, b1;
    b0.x = m0 * xc[(long)c0i * SEG_D];
    b0.y = m1 * xc[(long)c1i * SEG_D];
    b1.x = m2 * xc[(long)c2i * SEG_D];
    b1.y = m3 * xc[(long)c3i * SEG_D];

    acc0 = __builtin_amdgcn_wmma_f32_16x16x4_f32(
        false, a, false, b0, (short)0, acc0, false, false);
    acc1 = __builtin_amdgcn_wmma_f32_16x16x4_f32(
        false, a, false, b1, (short)0, acc1, false, false);
  }

  // Every row of D carries identical column sums; lane<16, VGPR0 has col's
  // sum. Exact f32 division matches the reference.
  float sum = acc0[0] + acc1[0];
  if (lane < 16) {
    out[(long)b * SEG_D + col] = sum / (float)size;
  }
}

// ---------------------------------------------------------------------------
// Host-side launcher. d_in[0] = x (float32, N*D), d_in[1] = sizes (int, B).
// Output: [B, D] float32. d_ws holds the (B+1) int offsets.
// ---------------------------------------------------------------------------
extern "C" void kernel_launch(void* const* d_in, const int* in_sizes, int n_in,
                              void* d_out, int out_size, void* d_ws, size_t ws_size,
                              hipStream_t stream) {
  const float* x     = (const float*)d_in[0];
  const int*   sizes = (const int*)d_in[1];
  float*       out   = (float*)d_out;

  const int B = in_sizes[1];        // 4096 segments
  int* offsets = (int*)d_ws;        // B+1 ints of scratch

  seg_offsets_kernel<<<1, 1024, 0, stream>>>(sizes, offsets, B);
  seg_mean_wmma_kernel<<<B, 512, 0, stream>>>(x, sizes, offsets, out);
}